// VanillaRecursiveNN_25589415149692
// MI455X (gfx1250) — compile-verified
//
#include <hip/hip_runtime.h>
#include <hip/hip_bf16.h>

typedef float v2f __attribute__((ext_vector_type(2)));
typedef float v8f __attribute__((ext_vector_type(8)));
typedef int   b128_t __attribute__((vector_size(16)));   // matches builtin param type

#define BM 128
#define BN 64
#define BK 32
#define LDA 34   // padded LDS stride (dwords): m*34 mod 64 distinct for m=0..15
#define LDB 34

#if __has_builtin(__builtin_amdgcn_global_load_async_to_lds_b128) && \
    __has_builtin(__builtin_amdgcn_s_wait_asynccnt)
#define HAVE_ASYNC 1
#else
#define HAVE_ASYNC 0
#endif

// Unconditional 16B stage: global -> LDS via CDNA5 async DMA (ASYNCcnt).
__device__ __forceinline__ void stage_full16B(const float* __restrict__ src, float* dst) {
#if HAVE_ASYNC
    __builtin_amdgcn_global_load_async_to_lds_b128(
        (__attribute__((address_space(1))) b128_t*)src,
        (__attribute__((address_space(3))) b128_t*)dst,
        /*offset=*/0, /*cpol=*/0);
#else
    const float4 v = *(const float4*)src;
    dst[0] = v.x; dst[1] = v.y; dst[2] = v.z; dst[3] = v.w;
#endif
}

// Guarded 16B stage with zero-fill past K (tail chunk only).
__device__ __forceinline__ void stage_guard16B(const float* __restrict__ src,
                                               float* dst, int krel, int K) {
#pragma unroll
    for (int c = 0; c < 4; ++c) {
        const int k = krel + c;
        dst[c] = (k < K) ? src[k] : 0.0f;
    }
}

// Generic fp32 WMMA GEMM:  C[M,N] = X[M,K] * W[N,K]^T (+ bias)
// Optional row gather: if ids != nullptr, row m of X is emb[ids[m]] (length K).
// Requires: M % 128 == 0, N % 64 == 0, K >= 32. K zero-padded in LDS.
__global__ __launch_bounds__(256)
void wmma_gemm_f32(const float* __restrict__ X,      // [M,K] row-major (or null if gather)
                   const float* __restrict__ W,      // [N,K] row-major
                   const float* __restrict__ bias,   // [N] or null
                   float* __restrict__ C,            // [M,N] row-major
                   const int* __restrict__ ids,      // [M] gather indices or null
                   const float* __restrict__ emb,    // gather table [*,K] or null
                   int M, int N, int K)
{
    __shared__ float As[2][BM * LDA];   // As[m][k]
    __shared__ float Bs[2][BN * LDB];   // Bs[n][k]  (== B-matrix[k][n])

    const int tid   = threadIdx.x;
    const int lane  = tid & 31;
    const int wave  = tid >> 5;           // 0..7
    const int tileM = blockIdx.x * BM;
    const int tileN = blockIdx.y * BN;

    const int lhalf = lane >> 4;          // 0/1: selects K pair in A/B frags
    const int lrow  = lane & 15;          // M row (A frag) / N col (B frag)

    // ---- hoisted per-thread staging sources/dests (K-loop invariant) ----
    // 8 threads per row, 4 floats each: A = 4 passes of 32 rows, B = 2 passes.
    const int colb = (tid & 7) * 4;       // 0..28
    const int rA   = tid >> 3;            // 0..31
    const int ASTRIDE = BM * LDA;
    const int BSTRIDE = BN * LDB;

    const float* sA[4]; float* dA[4];
#pragma unroll
    for (int g = 0; g < 4; ++g) {
        const int row = tileM + g * 32 + rA;
        sA[g] = ids ? (emb + (size_t)ids[row] * K) : (X + (size_t)row * K);
        dA[g] = &As[0][(g * 32 + rA) * LDA + colb];
    }
    const float* sB[2]; float* dB[2];
#pragma unroll
    for (int g = 0; g < 2; ++g) {
        const int row = tileN + g * 32 + rA;
        sB[g] = W + (size_t)row * K;
        dB[g] = &Bs[0][(g * 32 + rA) * LDB + colb];
    }

    v8f acc[4];
    acc[0] = (v8f)0.0f; acc[1] = (v8f)0.0f; acc[2] = (v8f)0.0f; acc[3] = (v8f)0.0f;

    const int nfull   = K / BK;                   // branch-free async chunks
    const int rem     = K - nfull * BK;           // tail columns (leaf: 300-288=12)
    const int nchunks = nfull + (rem ? 1 : 0);

    // prologue: stage chunk 0 into buffer 0 (K >= 32 so chunk 0 is full)
#pragma unroll
    for (int g = 0; g < 4; ++g) stage_full16B(sA[g] + colb, dA[g]);
#pragma unroll
    for (int g = 0; g < 2; ++g) stage_full16B(sB[g] + colb, dB[g]);

    int buf = 0;
    for (int c = 0; c < nchunks; ++c) {
#if HAVE_ASYNC
        __builtin_amdgcn_s_wait_asynccnt(0);   // our async fills of [buf] done
#endif
        __syncthreads();                        // everyone's fills of [buf] visible

        // stage chunk c+1 into the other buffer while computing on [buf]
        const int nx = c + 1;
        const int o  = buf ^ 1;
        if (nx < nfull) {                       // uniform branch, no exec churn
            const int krel = nx * BK + colb;
#pragma unroll
            for (int g = 0; g < 4; ++g) stage_full16B(sA[g] + krel, dA[g] + o * ASTRIDE);
#pragma unroll
            for (int g = 0; g < 2; ++g) stage_full16B(sB[g] + krel, dB[g] + o * BSTRIDE);
        } else if (nx < nchunks) {              // single partial tail chunk
            const int krel = nfull * BK + colb;
#pragma unroll
            for (int g = 0; g < 4; ++g) stage_guard16B(sA[g], dA[g] + o * ASTRIDE, krel, K);
#pragma unroll
            for (int g = 0; g < 2; ++g) stage_guard16B(sB[g], dB[g] + o * BSTRIDE, krel, K);
        }

        // ---- compute: 8 k-steps of 16x16x4; wave owns 16(M) x 64(N) ----
        const float* aBase = &As[buf][(wave * 16 + lrow) * LDA];
        const float* bBase = &Bs[buf][lrow * LDB];
#pragma unroll
        for (int kk = 0; kk < BK; kk += 4) {
            const int kb = kk + lhalf * 2;
            const v2f a = *(const v2f*)(aBase + kb);
#pragma unroll
            for (int nt = 0; nt < 4; ++nt) {
                const v2f b = *(const v2f*)(bBase + nt * 16 * LDB + kb);
                acc[nt] = __builtin_amdgcn_wmma_f32_16x16x4_f32(
                    /*neg_a=*/false, a, /*neg_b=*/false, b,
                    /*c_mod=*/(short)0, acc[nt],
                    /*reuse_a=*/false, /*reuse_b=*/false);
            }
        }
        buf ^= 1;
    }

    // ---- epilogue: optional bias add, store 16x16 tiles ----
    // C/D layout: vgpr i -> row (i + lhalf*8), col = lrow
#pragma unroll
    for (int nt = 0; nt < 4; ++nt) {
        const int n   = tileN + nt * 16 + lrow;
        const float bv = bias ? bias[n] : 0.0f;
#pragma unroll
        for (int i = 0; i < 8; ++i) {
            const int m = tileM + wave * 16 + i + lhalf * 8;
            C[(size_t)m * N + n] = acc[nt][i] + bv;
        }
    }
}

extern "C" void kernel_launch(void* const* d_in, const int* in_sizes, int n_in,
                              void* d_out, int out_size, void* d_ws, size_t ws_size,
                              hipStream_t stream) {
    (void)in_sizes; (void)n_in; (void)out_size; (void)ws_size;

    const int B_TREES = 256;
    const int LEAVES  = 256;
    const int WDIM    = 300;
    const int HID     = 512;

    const int*   word_ids = (const int*)  d_in[0];   // [B, L]
    const float* emb      = (const float*)d_in[1];   // [VOCAB, 300]
    const float* W1       = (const float*)d_in[2];   // [512, 300]
    const float* W2       = (const float*)d_in[3];   // [512, 1024]
    const float* b2       = (const float*)d_in[4];   // [512]
    float*       out      = (float*)d_out;           // [256, 512]

    float* buf0 = (float*)d_ws;                               // up to 65536 x 512
    float* buf1 = buf0 + (size_t)B_TREES * LEAVES * HID;      // up to 32768 x 512

    // ---- leaves: h0[m,n] = sum_d emb[ids[m],d] * W1[n,d] ----
    {
        const int M = B_TREES * LEAVES;     // 65536
        dim3 grid(M / BM, HID / BN);
        wmma_gemm_f32<<<grid, 256, 0, stream>>>(
            nullptr, W1, nullptr, buf0, word_ids, emb, M, HID, WDIM);
    }

    // ---- 8 tree levels: view [2M,512] buffer as [M,1024], GEMM with W2 + b2 ----
    float* cur = buf0;
    float* nxt = buf1;
    int n = LEAVES;
    for (int d = 0; d < 8; ++d) {
        const int M = B_TREES * (n / 2);           // 32768 ... 256
        float* dst = (d == 7) ? out : nxt;
        dim3 grid(M / BM, HID / BN);
        wmma_gemm_f32<<<grid, 256, 0, stream>>>(
            cur, W2, b2, dst, nullptr, nullptr, M, HID, 2 * HID);
        float* t = cur; cur = dst; nxt = t;
        n /= 2;
    }
}